// GraphConvSparse_49151605735712
// MI455X (gfx1250) — compile-verified
//
#include <hip/hip_runtime.h>
#include <hip/hip_bf16.h>
#include <stdint.h>

typedef __attribute__((ext_vector_type(2))) float v2f;
typedef __attribute__((ext_vector_type(8))) float v8f;

#define D 128   // D_IN == D_OUT == 128

// ---------------------------------------------------------------------------
// Kernel 1: zero the adjacency bitmap (uint4 stores, 32 MB)
// ---------------------------------------------------------------------------
__global__ void zero_bitmap_kernel(uint4* __restrict__ p, long n) {
    long i = (long)blockIdx.x * blockDim.x + threadIdx.x;
    if (i < n) p[i] = make_uint4(0u, 0u, 0u, 0u);
}

// ---------------------------------------------------------------------------
// Kernel 2: build symmetric 0/1 adjacency bitmap with set semantics.
// atomicOr naturally dedups duplicate edges (matches .set(1.0)).
// ---------------------------------------------------------------------------
__global__ void build_adj_kernel(const int* __restrict__ ei, uint32_t* __restrict__ bm,
                                 int E, int mask, int wpr) {
    int e = blockIdx.x * blockDim.x + threadIdx.x;
    if (e >= E) return;
    int r = ei[e]     & mask;
    int c = ei[E + e] & mask;
    atomicOr(&bm[(size_t)r * wpr + (c >> 5)], 1u << (c & 31));
    atomicOr(&bm[(size_t)c * wpr + (r >> 5)], 1u << (r & 31));
}

// ---------------------------------------------------------------------------
// Kernel 3: h = x @ W via V_WMMA_F32_16X16X4_F32 (native fp32 matrix core).
// One wave per 16x16 output tile; 8 waves/block cover all 8 N-tiles of D=128.
// A(16x4 f32): lane = (half<<4)|m, VGPR v holds K = k0 + v + 2*half.
// B(4x16 f32): lane = (half<<4)|n, VGPR v holds K = k0 + v + 2*half.
// C(16x16 f32): VGPR v, lane -> M = v + 8*half, N = lane&15.
// ---------------------------------------------------------------------------
__global__ __launch_bounds__(256) void gemm_xw_wmma_kernel(const float* __restrict__ x,
                                                           const float* __restrict__ w,
                                                           float* __restrict__ h) {
    const int lane = threadIdx.x & 31;
    const int wave = threadIdx.x >> 5;   // 0..7 -> N-tile
    const int half = lane >> 4;          // 0 or 1
    const int l16  = lane & 15;
    const int m0   = blockIdx.x * 16;
    const int n0   = wave * 16;

    v8f c = {};
    const float* __restrict__ arow = x + (size_t)(m0 + l16) * D;

    #pragma unroll 4
    for (int k = 0; k < D; k += 4) {
        const int ka = k + 2 * half;
        v2f a, b;
        a.x = arow[ka];
        a.y = arow[ka + 1];
        b.x = w[(size_t)ka       * D + n0 + l16];
        b.y = w[(size_t)(ka + 1) * D + n0 + l16];
        // 8 args: (neg_a, A, neg_b, B, c_mod, C, reuse_a, reuse_b)
        c = __builtin_amdgcn_wmma_f32_16x16x4_f32(false, a, false, b,
                                                  (short)0, c, false, false);
    }

    #pragma unroll
    for (int v = 0; v < 8; ++v) {
        h[(size_t)(m0 + v + 8 * half) * D + n0 + l16] = c[v];
    }
}

// ---------------------------------------------------------------------------
// Kernel 4: out[row,:] = relu( sum_{j in adj(row)} h[j,:] ).
// One 128-thread block per row; thread t owns column t. Bitmap addresses are
// workgroup-uniform -> scalar loads; each h[j,:] gather is a coalesced 512 B
// read (L2-resident: h is 8 MB, bitmap 32 MB, both << 192 MB L2).
// ---------------------------------------------------------------------------
__global__ __launch_bounds__(128) void aggregate_relu_kernel(const uint32_t* __restrict__ bm,
                                                             const float* __restrict__ h,
                                                             float* __restrict__ out,
                                                             int wpr) {
    const int row = blockIdx.x;
    const int t   = threadIdx.x;   // 0..127
    const uint4* __restrict__ rp = (const uint4*)(bm + (size_t)row * wpr);
    const int q = wpr >> 2;        // uint4 words per row

    float acc = 0.0f;
    for (int i = 0; i < q; ++i) {
        uint4 bits = rp[i];
        const int base = i << 7;   // i * 128 columns
        uint32_t b;
        b = bits.x; while (b) { int j = base +      __builtin_ctz(b); b &= b - 1; acc += h[(size_t)j * D + t]; }
        b = bits.y; while (b) { int j = base + 32 + __builtin_ctz(b); b &= b - 1; acc += h[(size_t)j * D + t]; }
        b = bits.z; while (b) { int j = base + 64 + __builtin_ctz(b); b &= b - 1; acc += h[(size_t)j * D + t]; }
        b = bits.w; while (b) { int j = base + 96 + __builtin_ctz(b); b &= b - 1; acc += h[(size_t)j * D + t]; }
    }
    out[(size_t)row * D + t] = fmaxf(acc, 0.0f);
}

// ---------------------------------------------------------------------------
// Launcher. Inputs: x[N*128] f32, edge_index[2*E] int, num_nodes[1], W[128*128] f32.
// Workspace: [0, N*N/8) adjacency bitmap, then [.., +N*128*4) h.
// ---------------------------------------------------------------------------
extern "C" void kernel_launch(void* const* d_in, const int* in_sizes, int n_in,
                              void* d_out, int out_size, void* d_ws, size_t ws_size,
                              hipStream_t stream) {
    const float* x  = (const float*)d_in[0];
    const int*   ei = (const int*)d_in[1];
    const float* w  = (const float*)d_in[3];
    float*       out = (float*)d_out;

    const int N   = in_sizes[0] / D;   // 16384
    const int E   = in_sizes[1] / 2;   // 524288
    const int wpr = N >> 5;            // 512 words per bitmap row

    uint32_t* bm = (uint32_t*)d_ws;
    const size_t bm_bytes = (size_t)N * (size_t)wpr * 4u;   // 32 MB
    float* h = (float*)((char*)d_ws + bm_bytes);            // 8 MB

    const long zero_n = (long)(bm_bytes / 16u);             // uint4 count
    zero_bitmap_kernel<<<(int)((zero_n + 255) / 256), 256, 0, stream>>>((uint4*)bm, zero_n);
    build_adj_kernel<<<(E + 255) / 256, 256, 0, stream>>>(ei, bm, E, N - 1, wpr);
    gemm_xw_wmma_kernel<<<N / 16, 256, 0, stream>>>(x, w, h);
    aggregate_relu_kernel<<<N, D, 0, stream>>>(bm, h, out, wpr);
}